// Network_12403865551324
// MI455X (gfx1250) — compile-verified
//
#include <hip/hip_runtime.h>

typedef float v2f __attribute__((ext_vector_type(2)));
typedef float v8f __attribute__((ext_vector_type(8)));

#define NPTS 262144
#define DIN  32
#define DOUT 32
#define KTAP 9

// out[i] = bias[i % 32]  (row-major [N, DOUT])
__global__ __launch_bounds__(256) void spconv_bias_init(const float* __restrict__ bias,
                                                        float* __restrict__ out,
                                                        int n_total) {
    int i = blockIdx.x * blockDim.x + threadIdx.x;
    if (i < n_total) out[i] = bias[i & (DOUT - 1)];
}

// One wave = one tile of 16 pairs for kernel-offset k = blockIdx.y.
// D[16x16] = gather(feat)[16x32] @ W_k[32x16] (out cols 0..15), scatter-added via f32 atomics.
__global__ __launch_bounds__(256) void spconv_wmma_f32(
    const float* __restrict__ feat,
    const float* __restrict__ weight,        // [DOUT, 3, 3, DIN]
    const int*   __restrict__ indice_pairs,  // [2, K, N]
    const int*   __restrict__ indice_pair_num,
    float*       __restrict__ out) {
    const int k    = blockIdx.y;
    const int n_k  = indice_pair_num[k];
    const int lane = threadIdx.x & 31;
    const int wave = threadIdx.x >> 5;
    const int p0   = (blockIdx.x * 8 + wave) * 16;
    if (p0 >= n_k) return;                       // wave-uniform: EXEC stays full

    const int half = lane >> 4;                  // 0: K+0/K+1, 1: K+2/K+3
    const int l16  = lane & 15;
    const int koff = half * 2;

    // A-matrix gather: row M = l16 of the 16-pair tile
    const int  prow   = p0 + l16;
    const bool rvalid = prow < n_k;
    const int  in_row = rvalid ? indice_pairs[k * NPTS + prow] : 0;
    const float* arow = feat + (long)in_row * DIN + koff;

    // B-matrix: W_k[c][n] = weight[(n*9 + k)*32 + c]; N = l16
    const float* brow = weight + (l16 * KTAP + k) * DIN + koff;

    v8f acc = {};
#pragma unroll
    for (int c = 0; c < 8; ++c) {                // K = 32 = 8 chunks of 4
        v2f a = *(const v2f*)(arow + 4 * c);
        if (!rvalid) { a.x = 0.0f; a.y = 0.0f; } // branchless zero for padded rows
        v2f b = *(const v2f*)(brow + 4 * c);
        // 8 args: (neg_a, A, neg_b, B, c_mod, C, reuse_a, reuse_b)
        acc = __builtin_amdgcn_wmma_f32_16x16x4_f32(
            false, a, false, b, (short)0, acc, false, false);
    }

    // C/D layout: VGPR v -> row p0 + v + 8*half, col l16
#pragma unroll
    for (int v = 0; v < 8; ++v) {
        const int srow = p0 + v + half * 8;
        if (srow < n_k) {
            const int orow = indice_pairs[(KTAP + k) * NPTS + srow];
            __hip_atomic_fetch_add(out + (long)orow * DOUT + l16, acc[v],
                                   __ATOMIC_RELAXED, __HIP_MEMORY_SCOPE_AGENT);
        }
    }
}

extern "C" void kernel_launch(void* const* d_in, const int* in_sizes, int n_in,
                              void* d_out, int out_size, void* d_ws, size_t ws_size,
                              hipStream_t stream) {
    const float* feat   = (const float*)d_in[0];
    const float* weight = (const float*)d_in[1];
    const float* bias   = (const float*)d_in[2];
    const int*   pairs  = (const int*)d_in[3];
    const int*   pnum   = (const int*)d_in[4];
    float* out = (float*)d_out;

    const int n_total = NPTS * DOUT;
    spconv_bias_init<<<(n_total + 255) / 256, 256, 0, stream>>>(bias, out, n_total);

    dim3 grid((NPTS + 127) / 128, KTAP);         // 8 waves/block * 16 pairs/wave
    spconv_wmma_f32<<<grid, 256, 0, stream>>>(feat, weight, pairs, pnum, out);
}